// NeuralSplineTransformer_25031069401607
// MI455X (gfx1250) — compile-verified
//
#include <hip/hip_runtime.h>
#include <math.h>
#include <stdint.h>

#define B_TOT   2048
#define F_TOT   512
#define KBINS   32
#define NP      (3*KBINS + 1)          // 97 parameter rows
#define F_TILE  256
#define THREADS 256
#define NVEC    ((NP * F_TILE) / 4)    // 6208 float4 transfers per block

__global__ void zero_logdet_kernel(float* __restrict__ ld) {
    int i = blockIdx.x * blockDim.x + threadIdx.x;
    if (i < B_TOT) ld[i] = 0.0f;
}

__device__ __forceinline__ float softplus_f(float v) {
    // numerically stable softplus: max(v,0) + log1p(exp(-|v|))
    return fmaxf(v, 0.0f) + log1pf(expf(-fabsf(v)));
}

__global__ __launch_bounds__(THREADS)
void rqs_spline_kernel(const float* __restrict__ x,
                       const float* __restrict__ params,
                       const float* __restrict__ x0,
                       const float* __restrict__ xf,
                       float* __restrict__ out_y,
                       float* __restrict__ out_ld)
{
    __shared__ float lds_p[NP * F_TILE];      // 97 x 256 floats = 99,328 B (3 blocks/WGP)
    __shared__ float wred[THREADS / 32];

    const int tid = threadIdx.x;
    const int b   = blockIdx.y;
    const int f0  = blockIdx.x * F_TILE;

    // ---- async stage: parameters[b, :, f0:f0+F_TILE] -> LDS ----
    // Stream-once data (407 MB total, no reuse, > 192 MB L2) -> non-temporal hint.
    {
        const uint64_t gbase = (uint64_t)(uintptr_t)(params + (size_t)b * NP * F_TOT + f0);
        const uint32_t lbase = (uint32_t)(uintptr_t)(&lds_p[0]);   // flat low 32 bits == LDS offset
        for (int i = tid; i < NVEC; i += THREADS) {
            uint32_t row  = (uint32_t)i >> 6;                 // 64 float4 per 256-float row
            uint32_t col  = (uint32_t)i & 63u;
            uint32_t goff = row * (F_TOT * 4u) + col * 16u;   // global byte offset (2048 B row stride)
            uint32_t loff = lbase + (uint32_t)i * 16u;        // dense LDS byte offset
            asm volatile("global_load_async_to_lds_b128 %0, %1, %2 th:TH_LOAD_NT"
                         :
                         : "v"(loff), "v"(goff), "s"(gbase)
                         : "memory");
        }
        asm volatile("s_wait_asynccnt 0" ::: "memory");
    }
    __syncthreads();

    const int   f    = f0 + tid;
    const float xv   = x[(size_t)b * F_TOT + f];
    const float x0v  = x0[f];
    const float span = xf[f] - x0v;

    // ---- pass 1: softmax maxima (stride 256 == 0 mod 64 banks -> conflict-free) ----
    float wmax = -INFINITY, hmax = -INFINITY;
    #pragma unroll 4
    for (int j = 0; j < KBINS; ++j) {
        wmax = fmaxf(wmax, lds_p[j * F_TILE + tid]);
        hmax = fmaxf(hmax, lds_p[(KBINS + j) * F_TILE + tid]);
    }

    // ---- pass 2: softmax denominators; cache exp() back into the same LDS slots.
    // Each thread touches only its own column (j*F_TILE + tid), so no barrier is
    // needed, and pass 3 avoids re-evaluating 64 quarter-rate transcendentals.
    float wsum = 0.0f, hsum = 0.0f;
    #pragma unroll 4
    for (int j = 0; j < KBINS; ++j) {
        float ew = expf(lds_p[j * F_TILE + tid] - wmax);
        float eh = expf(lds_p[(KBINS + j) * F_TILE + tid] - hmax);
        lds_p[j * F_TILE + tid]           = ew;
        lds_p[(KBINS + j) * F_TILE + tid] = eh;
        wsum += ew;
        hsum += eh;
    }
    const float wscale = span / wsum;
    const float hscale = span / hsum;

    // ---- pass 3: cumsum + monotone bin capture ----
    // knots_x[j] (left edge of bin j) is increasing; "xv > edge" is a prefix
    // predicate, so keeping the LAST j where it holds gives
    // bin = clip(sum(xv > knots_x) - 1, 0, KBINS-1) exactly.
    float cx = 0.0f, cy = 0.0f;
    float xk = x0v, yk = x0v, wb = 0.0f, hb = 0.0f;
    int bin = 0;
    #pragma unroll 4
    for (int j = 0; j < KBINS; ++j) {
        float wj   = lds_p[j * F_TILE + tid] * wscale;
        float hj   = lds_p[(KBINS + j) * F_TILE + tid] * hscale;
        float left = x0v + cx;
        if ((xv > left) || (j == 0)) {   // j==0 branch implements the low clip
            bin = j; xk = left; yk = x0v + cy; wb = wj; hb = hj;
        }
        cx += wj; cy += hj;
    }

    // ---- slope gather (data-dependent row, still conflict-free) + softplus ----
    const float dk  = softplus_f(lds_p[(2 * KBINS + bin    ) * F_TILE + tid]);
    const float dk1 = softplus_f(lds_p[(2 * KBINS + bin + 1) * F_TILE + tid]);

    // ---- rational-quadratic spline ----
    const float s    = hb / wb;
    const float eps  = (xv - xk) / wb;
    const float ome  = 1.0f - eps;
    const float e1m  = eps * ome;
    const float e2   = eps * eps;
    const float den  = s + (dk1 + dk - 2.0f * s) * e1m;
    const float y    = yk + hb * (s * e2 + dk * e1m) / den;
    const float dydx = s * s * (dk1 * e2 + 2.0f * s * e1m + dk * ome * ome) / (den * den);

    out_y[(size_t)b * F_TOT + f] = y;

    // ---- log-det reduction: wave32 shuffle tree -> per-wave -> one atomic ----
    float ld = logf(dydx);
    for (int off = 16; off > 0; off >>= 1)
        ld += __shfl_xor(ld, off, 32);
    const int wid = tid >> 5, lane = tid & 31;
    if (lane == 0) wred[wid] = ld;
    __syncthreads();
    if (tid == 0) {
        float t = 0.0f;
        #pragma unroll
        for (int w = 0; w < THREADS / 32; ++w) t += wred[w];
        atomicAdd(&out_ld[b], t);
    }
}

extern "C" void kernel_launch(void* const* d_in, const int* in_sizes, int n_in,
                              void* d_out, int out_size, void* d_ws, size_t ws_size,
                              hipStream_t stream) {
    (void)in_sizes; (void)n_in; (void)out_size; (void)d_ws; (void)ws_size;
    const float* x      = (const float*)d_in[0];
    const float* params = (const float*)d_in[1];
    const float* x0     = (const float*)d_in[2];
    const float* xf     = (const float*)d_in[3];

    float* out_y  = (float*)d_out;                       // (B, F) flattened
    float* out_ld = out_y + (size_t)B_TOT * F_TOT;       // (B,) log_det_J

    zero_logdet_kernel<<<(B_TOT + 255) / 256, 256, 0, stream>>>(out_ld);

    dim3 grid(F_TOT / F_TILE, B_TOT);                    // (2, 2048)
    rqs_spline_kernel<<<grid, THREADS, 0, stream>>>(x, params, x0, xf, out_y, out_ld);
}